// OnlinePhraseBuilder_28338194219453
// MI455X (gfx1250) — compile-verified
//
#include <hip/hip_runtime.h>
#include <hip/hip_bf16.h>

// ---------------------------------------------------------------------------
// OnlinePhraseBuilder for MI455X (gfx1250, wave32).
//
// Roofline: outputs = 136 MB of f32 -> ~5.9us floor at 23.3 TB/s. All other
// traffic (2MB ids, 2x200KB tables, 130KB bitmask scratch) is L2-resident
// noise. Zero FP FLOPs -> WMMA inapplicable; this is a fill + sparse-scatter
// problem with a short sequential recurrence resolved with wave32 bit tricks.
// ---------------------------------------------------------------------------

#define BDIM 64
#define TDIM 8192
#define NCHUNK (TDIM / 32) // 256 32-token chunks per row

// ---------------------------------------------------------------------------
// Stage 1: bulk default-fill of d_out with 128-bit stores.
// Two branch-free grid-stride loops (no per-element value select, which
// previously forced a scratch_store_b128 + pointer-cndmask dance):
//   [0, b4)  -> 0.0f   (phrase_mask + phrase_token_idx)
//   [b4, n4) -> -1.0f  (phrase_end_pos)
// Each loop body is address-add + global_store_b128, HBM-bound.
// ---------------------------------------------------------------------------
__global__ __launch_bounds__(256) void fill_out_kernel(float4* __restrict__ out,
                                                       long n4, long b4) {
    const long tid    = (long)blockIdx.x * blockDim.x + threadIdx.x;
    const long stride = (long)gridDim.x * blockDim.x;

    const float4 z = make_float4(0.f, 0.f, 0.f, 0.f);
    // zeros region, 2x unrolled
    long i = tid * 2;
    const long step = stride * 2;
    for (; i + 1 < b4; i += step) {
        out[i]     = z;
        out[i + 1] = z;
    }
    if (i < b4) out[i] = z;

    const float4 m = make_float4(-1.f, -1.f, -1.f, -1.f);
    // -1 region (phrase_end_pos defaults)
    for (long j = b4 + tid; j < n4; j += stride) out[j] = m;
}

__global__ void init_ws_kernel(int* __restrict__ lastReal) {
    lastReal[threadIdx.x] = -1;
}

// ---------------------------------------------------------------------------
// Stage 2: fully parallel per-token classification. Each wave32 compresses 32
// tokens into one e-bit word and one real-bit word via v_cmp + ballot, and
// contributes a wave-reduced atomicMax for last_real. Gather latency is hidden
// by 16K concurrent waves.
// ---------------------------------------------------------------------------
__global__ __launch_bounds__(256) void build_masks_kernel(
    const int* __restrict__ ids,
    const int* __restrict__ is_punct,   // bool table, harness-normalized ints
    const int* __restrict__ is_abbr,
    const int* __restrict__ p_pad,
    int* __restrict__ lastReal,
    unsigned* __restrict__ emask,
    unsigned* __restrict__ realmask) {
    const long g = (long)blockIdx.x * blockDim.x + threadIdx.x;
    const long total = (long)BDIM * TDIM;
    if (g >= total) return;
    const int b   = (int)(g / TDIM);
    const int tt  = (int)(g % TDIM);
    const int pad = p_pad[0];
    const int id  = ids[g];
    const bool real = (id != pad);
    bool prevAbbr = false;
    if (tt != 0) prevAbbr = (is_abbr[ids[g - 1]] != 0); // prev_is_abbr[:,0]=False
    const bool e = (is_punct[id] != 0) && !prevAbbr && real;

    // wave32: low 32 bits of __ballot are the full wave mask
    const unsigned em = (unsigned)__ballot(e);
    const unsigned rm = (unsigned)__ballot(real);
    if ((threadIdx.x & 31) == 0) {
        emask[g >> 5]    = em;
        realmask[g >> 5] = rm;
    }
    // wave-reduce max(real position) before the atomic (64 rows x 256 atomics)
    int lm = real ? tt : -1;
    for (int off = 16; off > 0; off >>= 1) {
        const int o = __shfl_xor(lm, off, 32);
        lm = (o > lm) ? o : lm;
    }
    if ((threadIdx.x & 31) == 0) atomicMax(&lastReal[b], lm);
}

// ---------------------------------------------------------------------------
// Stage 3: one wave32 per batch row resolves the boundary recurrence.
// Per 32-token chunk: forced fires collapse to a tiny uniform loop (a forced
// boundary at L+ML lands in-chunk at most until the first fire resets it).
// Per-lane phrase_id = fires-before-me (popc of fireMask below lane),
// slot = distance to previous fire (clz of fireMask below lane).
// ---------------------------------------------------------------------------
__global__ __launch_bounds__(32) void resolve_kernel(
    const int* __restrict__ p_ml,
    const int* __restrict__ lastRealArr,
    const unsigned* __restrict__ emask,
    const unsigned* __restrict__ realmask,
    float* __restrict__ maskOut,   // (B, T, ML)
    float* __restrict__ idxOut,    // (B, T, ML)
    float* __restrict__ endOut) {  // (B, T)
    const int lane = threadIdx.x;
    const int row  = blockIdx.x;
    const int ML   = p_ml[0];
    const int lastReal = lastRealArr[row];

    const unsigned* eRow = emask    + (size_t)row * NCHUNK;
    const unsigned* rRow = realmask + (size_t)row * NCHUNK;
    const unsigned laneLt = lane ? (0xFFFFFFFFu >> (32 - lane)) : 0u;

    int L = -1;          // last boundary position (uniform)
    int phraseBase = 0;  // fires before current chunk (uniform)
    bool done = false;

    for (int sc = 0; sc < NCHUNK && !done; sc += 32) {
        // 32 chunk words per superchunk, coalesced; broadcast via shfl below.
        const unsigned eW = eRow[sc + lane];
        const unsigned rW = rRow[sc + lane];
        if (lane == 0 && sc + 32 < NCHUNK)
            __builtin_prefetch(eRow + sc + 32, 0, 0); // global_prefetch_b8

        for (int i = 0; i < 32; ++i) {
            const int t0 = (sc + i) << 5;
            if (t0 > lastReal) { done = true; break; } // no fires/reals beyond
            unsigned baseMask     = (unsigned)__shfl((int)eW, i, 32);
            const unsigned realM  = (unsigned)__shfl((int)rW, i, 32);
            // clamp fires to t <= lastReal and force a fire at t == lastReal
            const int k = lastReal - t0;
            if (k < 31)       { baseMask &= (2u << k) - 1u; baseMask |= 1u << k; }
            else if (k == 31) { baseMask |= 0x80000000u; }

            // uniform forced-fire resolution (<= #fires+1 iterations)
            unsigned fireMask = 0u;
            int Lc = L;
            while (true) {
                unsigned rem = baseMask;
                if (Lc >= t0) {
                    const int sh = Lc - t0;
                    rem = (sh >= 31) ? 0u : (rem & ~((2u << sh) - 1u));
                }
                const int nextE = rem ? (t0 + (__ffs((int)rem) - 1)) : 0x7FFFFFFF;
                const int cF = Lc + ML;                 // forced candidate
                const int f  = (cF < nextE) ? cF : nextE;
                if (f > lastReal || (unsigned)(f - t0) > 31u) break;
                fireMask |= 1u << (f - t0);
                Lc = f;
            }

            // per-lane outputs
            const int t = t0 + lane;
            const unsigned below = fireMask & laneLt;
            const int p     = phraseBase + __popc(below);
            const int prevB = below ? (t0 + (31 - __clz((int)below))) : L;
            const int slot  = t - prevB - 1;
            const bool real = (realM >> lane) & 1u;
            if (real && (unsigned)slot < (unsigned)ML) {
                const size_t o = ((size_t)row * TDIM + (size_t)p) * (size_t)ML
                               + (size_t)slot;
                maskOut[o] = 1.0f;
                idxOut[o]  = (float)t;
            }
            if ((fireMask >> lane) & 1u)
                endOut[(size_t)row * TDIM + (size_t)p] = (float)t;

            phraseBase += __popc(fireMask);
            L = Lc;
        }
    }
}

// ---------------------------------------------------------------------------
extern "C" void kernel_launch(void* const* d_in, const int* in_sizes, int n_in,
                              void* d_out, int out_size, void* d_ws, size_t ws_size,
                              hipStream_t stream) {
    const int* ids      = (const int*)d_in[0]; // (B,T) int32
    const int* is_punct = (const int*)d_in[1]; // (V,) bool->int
    const int* is_abbr  = (const int*)d_in[2]; // (V,) bool->int
    const int* p_ml     = (const int*)d_in[3]; // scalar max_phrase_len
    const int* p_pad    = (const int*)d_in[4]; // scalar pad_token_id

    const long N  = (long)in_sizes[0];             // B*T = 524288
    const long ML = ((long)out_size / N - 1) / 2;  // = 32 (host mirror of p_ml)

    float* out = (float*)d_out;
    float* maskOut = out;
    float* idxOut  = out + (size_t)(N * ML);
    float* endOut  = out + (size_t)(2 * N * ML);

    // workspace layout: [64 x int lastReal | pad to 256B | emask | realmask]
    int* lastReal      = (int*)d_ws;
    unsigned* emaskW   = (unsigned*)((char*)d_ws + 256);
    unsigned* realW    = emaskW + (N / 32);

    // 1) bulk fill defaults (HBM-bound, ~5.9us floor)
    const long n4 = (long)out_size / 4;
    const long b4 = (2 * N * ML) / 4;
    fill_out_kernel<<<4096, 256, 0, stream>>>((float4*)out, n4, b4);

    // 2) per-token bits + last_real
    init_ws_kernel<<<1, BDIM, 0, stream>>>(lastReal);
    build_masks_kernel<<<(int)(N / 256), 256, 0, stream>>>(
        ids, is_punct, is_abbr, p_pad, lastReal, emaskW, realW);

    // 3) per-row sequential resolve + sparse scatter
    resolve_kernel<<<BDIM, 32, 0, stream>>>(
        p_ml, lastReal, emaskW, realW, maskOut, idxOut, endOut);
}